// PolyConv_13314398617738
// MI455X (gfx1250) — compile-verified
//
#include <hip/hip_runtime.h>
#include <stdint.h>

#define TPB 256
#define WPB 8  // waves per block (wave32)

// ---------------------------------------------------------------------------
// CDNA5 async gather: global -> LDS, 16 bytes per lane, tracked on ASYNCcnt.
// lds_off is the wave-relative LDS byte address (low 32 bits of the generic
// pointer to a __shared__ element), gp is the per-lane global address.
// ---------------------------------------------------------------------------
__device__ __forceinline__ void async_gather_b128(uint32_t lds_off, const float* gp) {
  asm volatile("global_load_async_to_lds_b128 %0, %1, off"
               :: "v"(lds_off), "v"(gp) : "memory");
}

__global__ void zero_kernel(float* __restrict__ p, int n) {
  int i = blockIdx.x * blockDim.x + threadIdx.x;
  if (i < n) p[i] = 0.0f;
}

__global__ void deg_kernel(const int* __restrict__ src, int* __restrict__ deg, int E) {
  int e = blockIdx.x * blockDim.x + threadIdx.x;
  if (e < E) atomicAdd(&deg[src[e]], 1);
}

// w[e] = dinv_sqrt(src) * dinv_sqrt(dst); dinv_sqrt(i) = deg>0 ? rsqrt(deg) : 0
__global__ void weight_kernel(const int* __restrict__ src, const int* __restrict__ dst,
                              const int* __restrict__ deg, float* __restrict__ w, int E) {
  int e = blockIdx.x * blockDim.x + threadIdx.x;
  if (e < E) {
    int s = src[e], d = dst[e];
    int ds_i = deg[s], dd_i = deg[d];
    float a = (ds_i > 0) ? rsqrtf((float)ds_i) : 0.0f;
    float b = (dd_i > 0) ? rsqrtf((float)dd_i) : 0.0f;
    w[e] = a * b;
  }
}

// ---------------------------------------------------------------------------
// SpMM: y[dst[e]] += w[e] * x[src[e]]  (rows of 64 f32)
// Each wave handles 2 edges/iter: lanes 0-15 -> edge A, 16-31 -> edge B,
// each lane owns a float4 chunk of the row. The row gather for the NEXT pair
// is staged into per-wave LDS with global_load_async_to_lds_b128 while the
// current pair is multiplied and scattered with native f32 atomics.
// ---------------------------------------------------------------------------
__global__ __launch_bounds__(TPB) void spmm_kernel(
    const float* __restrict__ x, const int* __restrict__ src,
    const int* __restrict__ dst, const float* __restrict__ w,
    float* __restrict__ y, int nEdges) {
  // per wave: 2 buffers x 2 rows x 256B = 1KB -> 8KB per block
  __shared__ __align__(16) float lds[WPB * 256];

  const int lane = threadIdx.x & 31;
  const int wv   = threadIdx.x >> 5;
  const int g    = lane >> 4;   // which edge of the pair
  const int f    = lane & 15;   // float4 chunk within the row
  const int nPairs = (nEdges + 1) >> 1;
  const int nWaves = gridDim.x * WPB;

  // wave-relative LDS byte address of this lane's 16B write slot (parity 0)
  const uint32_t wr0 = (uint32_t)(uintptr_t)(&lds[wv * 256]) + (uint32_t)(lane * 16);

  int p = blockIdx.x * WPB + wv;

  // prologue: stage pair p into buffer 0
  if (p < nPairs) {
    int e = 2 * p + g;
    if (e < nEdges)
      async_gather_b128(wr0, x + (size_t)src[e] * 64 + f * 4);
  }

  int par = 0;
  for (; p < nPairs; p += nWaves) {
    int pn = p + nWaves;
    if (pn < nPairs) {
      // stage next pair into the other buffer, then wait for current only.
      // (async loads complete in order; pn<nPairs guarantees the stage
      //  instruction issues, so <=1 leaves exactly the new stage in flight)
      int e2 = 2 * pn + g;
      if (e2 < nEdges)
        async_gather_b128(wr0 + (uint32_t)((par ^ 1) * 512),
                          x + (size_t)src[e2] * 64 + f * 4);
      asm volatile("s_wait_asynccnt 0x1" ::: "memory");
    } else {
      asm volatile("s_wait_asynccnt 0x0" ::: "memory");
    }

    int e = 2 * p + g;
    if (e < nEdges) {
      float ww = w[e];
      int   d  = dst[e];
      const float* lp = &lds[wv * 256 + par * 128 + lane * 4];  // ds_load_b128
      float r0 = lp[0] * ww, r1 = lp[1] * ww, r2 = lp[2] * ww, r3 = lp[3] * ww;
      float* yp = y + (size_t)d * 64 + f * 4;
      // native fire-and-forget f32 atomics (STOREcnt; s_endpgm waits idle)
      asm volatile(
          "global_atomic_add_f32 %0, %1, off\n\t"
          "global_atomic_add_f32 %0, %2, off offset:4\n\t"
          "global_atomic_add_f32 %0, %3, off offset:8\n\t"
          "global_atomic_add_f32 %0, %4, off offset:12"
          :: "v"(yp), "v"(r0), "v"(r1), "v"(r2), "v"(r3) : "memory");
    }
    par ^= 1;
  }
}

__global__ void scale_kernel(const float4* __restrict__ x, float4* __restrict__ out,
                             float c, int n4) {
  int i = blockIdx.x * blockDim.x + threadIdx.x;
  if (i < n4) {
    float4 v = x[i];
    out[i] = make_float4(c * v.x, c * v.y, c * v.z, c * v.w);
  }
}

__global__ void axpy_kernel(const float4* __restrict__ x, float4* __restrict__ out,
                            float c, int n4) {
  int i = blockIdx.x * blockDim.x + threadIdx.x;
  if (i < n4) {
    float4 v = x[i];
    float4 o = out[i];
    out[i] = make_float4(o.x + c * v.x, o.y + c * v.y, o.z + c * v.z, o.w + c * v.w);
  }
}

static inline size_t align256(size_t v) { return (v + 255) & ~(size_t)255; }

extern "C" void kernel_launch(void* const* d_in, const int* in_sizes, int n_in,
                              void* d_out, int out_size, void* d_ws, size_t ws_size,
                              hipStream_t stream) {
  const float* feat = (const float*)d_in[0];
  const int*   src  = (const int*)d_in[1];
  const int*   dst  = (const int*)d_in[2];
  float*       out  = (float*)d_out;

  const int E  = in_sizes[1];
  const int NF = in_sizes[0];     // N * 64
  const int N  = NF / 64;
  const int N4 = NF / 4;

  // workspace layout (all 256B aligned)
  char* ws = (char*)d_ws;
  size_t o = 0;
  int*   deg = (int*)(ws + o);   o += align256((size_t)N * 4);
  float* w   = (float*)(ws + o); o += align256((size_t)E * 4);
  float* xA  = (float*)(ws + o); o += align256((size_t)NF * 4);
  float* xB  = (float*)(ws + o);

  const int bE  = (E + TPB - 1) / TPB;
  const int bN  = (N + TPB - 1) / TPB;
  const int bNF = (NF + TPB - 1) / TPB;
  const int bN4 = (N4 + TPB - 1) / TPB;
  const int bSp = 3200;  // 25600 waves -> ~16 edge-pairs per wave

  // degree + edge weights (shared across all 3 SpMM passes)
  zero_kernel<<<bN, TPB, 0, stream>>>((float*)deg, N);
  deg_kernel<<<bE, TPB, 0, stream>>>(src, deg, E);
  weight_kernel<<<bE, TPB, 0, stream>>>(src, dst, deg, w, E);

  // pass 1: xA = A @ feat ; h = (theta0+theta1) * xA
  zero_kernel<<<bNF, TPB, 0, stream>>>(xA, NF);
  spmm_kernel<<<bSp, TPB, 0, stream>>>(feat, src, dst, w, xA, E);
  scale_kernel<<<bN4, TPB, 0, stream>>>((const float4*)xA, (float4*)out, 0.8f, N4);

  // pass 2: xB = A @ xA ; h += theta2 * xB
  zero_kernel<<<bNF, TPB, 0, stream>>>(xB, NF);
  spmm_kernel<<<bSp, TPB, 0, stream>>>(xA, src, dst, w, xB, E);
  axpy_kernel<<<bN4, TPB, 0, stream>>>((const float4*)xB, (float4*)out, 0.15f, N4);

  // pass 3: xA = A @ xB ; h += theta3 * xA
  zero_kernel<<<bNF, TPB, 0, stream>>>(xA, NF);
  spmm_kernel<<<bSp, TPB, 0, stream>>>(xB, src, dst, w, xA, E);
  axpy_kernel<<<bN4, TPB, 0, stream>>>((const float4*)xA, (float4*)out, 0.05f, N4);
}